// RoIPoolingV2_80109730005435
// MI455X (gfx1250) — compile-verified
//
#include <hip/hip_runtime.h>
#include <math.h>

// RoI max-pool (OUT 7x7), features (B=4, C=256, H=64, W=64) fp32, rois (N,5) int32.
// Output (N, C, 7, 7) fp32.
//
// Strategy (MI455X / gfx1250):
//  - Memory-bound gather; feature map (16 MB) is L2-resident (192 MB L2).
//  - One workgroup per (roi, 4 channels); one wave32 per channel.
//  - Stage each ROI window (<=48x48 f32 = 9 KB) into LDS with ONE Tensor Data
//    Mover descriptor (tensor_load_to_lds, TENSORcnt) -- strided 2D DMA,
//    replacing ~676 per-lane async loads + address VALU per tile.
//  - Fallback: per-element global_load_async_to_lds_b32 (ASYNCcnt).
//  - Bin bounds use the same fp32 arithmetic as the JAX reference.

#define ROIP_OUT_H 7
#define ROIP_OUT_W 7
#define ROIP_NBINS (ROIP_OUT_H * ROIP_OUT_W)

static constexpr int kC = 256;
static constexpr int kH = 64;
static constexpr int kW = 64;
static constexpr int kChPerBlk = 4;          // one wave32 per channel
static constexpr int kMaxTile  = 48 * 48;    // max ROI window elements

typedef __attribute__((address_space(1))) int gint_t;   // global int*
typedef __attribute__((address_space(3))) int sint_t;   // LDS int*
typedef unsigned int v4u __attribute__((ext_vector_type(4)));
typedef int          v8i __attribute__((ext_vector_type(8)));
typedef int          v4i __attribute__((ext_vector_type(4)));

#define ROIP_HAS_TDM (__has_builtin(__builtin_amdgcn_tensor_load_to_lds) && \
                      __has_builtin(__builtin_amdgcn_s_wait_tensorcnt))

// Stage a roi_h x roi_w fp32 tile (row stride kW) from global `src` into LDS `lt`.
__device__ __forceinline__ void roip_stage_tile(const float* src, float* lt,
                                                int roi_w, int roi_h, int lane) {
#if ROIP_HAS_TDM
  const unsigned long long gaddr = (unsigned long long)(const void*)src;
  const unsigned int lds_off = (unsigned int)(unsigned long long)(void*)lt;

  v4u g0;
  g0[0] = 1u;                                        // count=1 valid user D#
  g0[1] = lds_off;                                   // lds_addr (bytes)
  g0[2] = (unsigned int)(gaddr & 0xFFFFFFFFu);       // global_addr[31:0]
  g0[3] = (unsigned int)((gaddr >> 32) & 0x01FFFFFFu)// global_addr[56:32]
          | (2u << 30);                              // type = 2 ("image")

  const unsigned int rw = (unsigned int)roi_w;
  const unsigned int rh = (unsigned int)roi_h;
  v8i g1;
  g1[0] = (int)(2u << 16);                 // wg_mask=0, data_size=2 (4B), flags=0
  g1[1] = (int)(rw << 16);                 // abar_addr=0 | tensor_dim0[15:0]
  g1[2] = (int)((rw >> 16) | (rh << 16));  // tensor_dim0[31:16] | tensor_dim1[15:0]
  g1[3] = (int)((rh >> 16) | (rw << 16));  // tensor_dim1[31:16] | tile_dim0
  g1[4] = (int)rh;                         // tile_dim1 | tile_dim2=0 (2D)
  g1[5] = kW;                              // tensor_dim0_stride[31:0] = 64 elems
  g1[6] = 0;                               // stride0[47:32]=0 | stride1[15:0]=0
  g1[7] = 0;                               // stride1[47:16]=0 (unused, 2D tile)

  v4i g2 = {0, 0, 0, 0};
  v4i g3 = {0, 0, 0, 0};
  v8i g4 = {0, 0, 0, 0, 0, 0, 0, 0};       // 6-arg toolchain: extra group, zeroed
  __builtin_amdgcn_tensor_load_to_lds(g0, g1, g2, g3, g4, /*cpol=*/0);
  __builtin_amdgcn_s_wait_tensorcnt(0);
#else
  for (int y = 0; y < roi_h; ++y) {
    const float* grow = src + (size_t)y * kW;
    float*       lrow = lt + y * roi_w;
    for (int x = lane; x < roi_w; x += 32) {
#if __has_builtin(__builtin_amdgcn_global_load_async_to_lds_b32)
      __builtin_amdgcn_global_load_async_to_lds_b32(
          (gint_t*)(unsigned long long)(const void*)(grow + x),
          (sint_t*)(unsigned int)(unsigned long long)(void*)(lrow + x),
          /*offset=*/0, /*cpol=*/0);
#else
      lrow[x] = grow[x];
#endif
    }
  }
#if __has_builtin(__builtin_amdgcn_s_wait_asynccnt)
  __builtin_amdgcn_s_wait_asynccnt(0);
#else
  asm volatile("s_wait_asynccnt 0" ::: "memory");
#endif
#endif
}

__global__ __launch_bounds__(kChPerBlk * 32)
void roipool_gfx1250_kernel(const float* __restrict__ feat,
                            const int*   __restrict__ rois,
                            float*       __restrict__ out) {
  __shared__ float tile[kChPerBlk][kMaxTile];

  const int wave = threadIdx.x >> 5;
  const int lane = threadIdx.x & 31;
  const int roi  = blockIdx.x;
  const int c    = blockIdx.y * kChPerBlk + wave;

  // ROI params (uniform across block -> scalar loads)
  const int b  = rois[roi * 5 + 0];
  const int x1 = rois[roi * 5 + 1];
  const int y1 = rois[roi * 5 + 2];
  const int x2 = rois[roi * 5 + 3];
  const int y2 = rois[roi * 5 + 4];
  const int roi_w = x2 - x1 + 1;   // in [5, 48]
  const int roi_h = y2 - y1 + 1;   // in [5, 48]

  // Base of this channel's ROI window in global memory (tile start).
  const float* src = feat + (((size_t)b * kC + c) * kH + y1) * (size_t)kW + x1;
  float* lt = tile[wave];          // this wave's private LDS slice

  roip_stage_tile(src, lt, roi_w, roi_h, lane);

  // --- Compute the 49 bins; fp32 math mirrors the reference exactly. ---
  const float bh = (float)roi_h / 7.0f;
  const float bw = (float)roi_w / 7.0f;

  for (int t = lane; t < ROIP_NBINS; t += 32) {
    const int i = t / ROIP_OUT_W;
    const int j = t - i * ROIP_OUT_W;

    int hs = (int)floorf((float)i       * bh);
    int he = (int)floorf((float)(i + 1) * bh);
    int ws = (int)floorf((float)j       * bw);
    int we = (int)floorf((float)(j + 1) * bw);
    hs = min(hs, roi_h - 1);  he = min(he, roi_h);
    ws = min(ws, roi_w - 1);  we = min(we, roi_w);

    const bool valid = (he > hs) && (we > ws);
    float m = -INFINITY;
    for (int y = hs; y < he; ++y) {
      const float* lrow = lt + y * roi_w;
      for (int x = ws; x < we; ++x) {
        m = fmaxf(m, lrow[x]);
      }
    }
    // Output layout (N, C, 7, 7); lanes of a wave write contiguous floats.
    out[((size_t)roi * kC + c) * ROIP_NBINS + t] = valid ? m : 0.0f;
  }
}

extern "C" void kernel_launch(void* const* d_in, const int* in_sizes, int n_in,
                              void* d_out, int out_size, void* d_ws, size_t ws_size,
                              hipStream_t stream) {
  const float* feat = (const float*)d_in[0];
  const int*   rois = (const int*)d_in[1];
  float*       out  = (float*)d_out;

  const int N = in_sizes[1] / 5;                 // number of ROIs
  dim3 grid((unsigned)N, (unsigned)(kC / kChPerBlk));
  dim3 block(kChPerBlk * 32);
  hipLaunchKernelGGL(roipool_gfx1250_kernel, grid, block, 0, stream,
                     feat, rois, out);
}